// LSTM_23613730193841
// MI455X (gfx1250) — compile-verified
//
#include <hip/hip_runtime.h>

// ---------------------------------------------------------------------------
// LSTM (IN=6, H=40, T=262144) for MI455X / gfx1250.
//  Kernel 1: x_proj = x @ W_ih^T + (b_ih+b_hh) via V_WMMA_F32_16X16X4_F32.
//  Kernel 2: single-workgroup 160-lane recurrent scan (latency-optimized).
//  Chunked so each x_proj tile (<=5.2MB) stays L2-resident between kernels.
// ---------------------------------------------------------------------------

typedef __attribute__((ext_vector_type(2))) float v2f;
typedef __attribute__((ext_vector_type(8))) float v8f;

#define LSTM_H 40
#define LSTM_G 160   // 4*H
#define LSTM_IN 6

// log2(e), 2*log2(e)
#define LOG2E  1.442695040888963f
#define LOG2E2 2.885390081777926f

__device__ __forceinline__ float fast_sigmoid(float z) {
    // sigma(z) = 1 / (1 + 2^(-z*log2e))   -> v_exp_f32 + v_rcp_f32
    float e = __builtin_amdgcn_exp2f(-z * LOG2E);
    return __builtin_amdgcn_rcpf(1.0f + e);
}
__device__ __forceinline__ float fast_tanh(float z) {
    // tanh(z) = 2 / (1 + 2^(-2z*log2e)) - 1
    float e = __builtin_amdgcn_exp2f(-z * LOG2E2);
    return 2.0f * __builtin_amdgcn_rcpf(1.0f + e) - 1.0f;
}

// ---------------------------------------------------------------------------
// Kernel 1: WMMA input-projection GEMM for one time-chunk.
//   xp[t][n] = sum_k x[t][k]*W_ih[n][k] + b_ih[n] + b_hh[n]
// One wave (32 threads) per 16(M=time) x 16(N=gate) tile. K=6 padded to 8 ->
// two V_WMMA_F32_16X16X4_F32, accumulator pre-seeded with the bias.
// All loads are unconditional & in-bounds; K=6,7 padding is done with
// branch-free v_cndmask selects (no exec-mask divergence).
// ---------------------------------------------------------------------------
__global__ __launch_bounds__(32) void xproj_wmma_kernel(
    const float* __restrict__ x,     // chunk base, [chunkT, 6]
    const float* __restrict__ W_ih,  // [160, 6]
    const float* __restrict__ b_ih,  // [160]
    const float* __restrict__ b_hh,  // [160]
    float* __restrict__ xp,          // [chunkT, 160]
    int chunkT)
{
    const int tile = blockIdx.x;                 // (chunkT/16) * 10 tiles
    const int mt = tile / 10;                    // time tile
    const int nt = tile - mt * 10;               // gate tile
    const int lane = threadIdx.x & 31;
    const int lm = lane & 15;                    // low lane index
    const int hi = lane >> 4;                    // lane half (0/1)

    const int row0 = mt * 16;                    // first timestep of tile
    const int n = nt * 16 + lm;                  // gate column for this lane

    // C/D layout: VGPR v -> (M = v + 8*hi, N = lm). Bias depends on N only.
    const float bias = b_ih[n] + b_hh[n];
    v8f acc;
#pragma unroll
    for (int v = 0; v < 8; ++v) acc[v] = bias;

    // A(16x4)/B(4x16) f32 layouts: K = kb + 2*hi + {0,1} for VGPR {0,1}.
    const float* xr = x + (size_t)(row0 + lm) * LSTM_IN;
    const float* wr = W_ih + (size_t)n * LSTM_IN;

    // --- WMMA #1: K = 0..3, every element valid.
    const int k0 = hi * 2;                       // 0 or 2
    v2f a0, b0;
    a0.x = xr[k0];     a0.y = xr[k0 + 1];
    b0.x = wr[k0];     b0.y = wr[k0 + 1];
    acc = __builtin_amdgcn_wmma_f32_16x16x4_f32(
        false, a0, false, b0, (short)0, acc, false, false);

    // --- WMMA #2: K = 4..7. hi=0 lanes carry K=4,5 (valid); hi=1 lanes
    // carry K=6,7 which must be zero. Load unconditionally from the valid
    // K=4,5 slots (always in-bounds) and select-zero on hi lanes.
    float xa = xr[4], xb = xr[5];
    float wa = wr[4], wb = wr[5];
    v2f a1, b1;
    a1.x = hi ? 0.0f : xa;   a1.y = hi ? 0.0f : xb;
    b1.x = hi ? 0.0f : wa;   b1.y = hi ? 0.0f : wb;
    acc = __builtin_amdgcn_wmma_f32_16x16x4_f32(
        false, a1, false, b1, (short)0, acc, false, false);

    // Store D: row = row0 + v + 8*hi, col = nt*16 + lm
    float* out = xp + (size_t)(row0 + hi * 8) * LSTM_G + nt * 16 + lm;
#pragma unroll
    for (int v = 0; v < 8; ++v) out[(size_t)v * LSTM_G] = acc[v];
}

// ---------------------------------------------------------------------------
// Kernel 2: sequential scan over one chunk. 160 lanes (5 wave32).
// Lane j owns gate row j: z_j = xp[t][j] + W_hh[j,:] . h  (W row in VGPRs,
// h broadcast from LDS via b128 loads, 4 split accumulators).
// Gate order (torch): i[0:40) f[40:80) g[80:120) o[120:160).
// ---------------------------------------------------------------------------
__global__ __launch_bounds__(LSTM_G) void lstm_scan_kernel(
    const float* __restrict__ xp,    // [chunkT, 160] (L2-resident chunk)
    const float* __restrict__ W_hh,  // [160, 40]
    float* __restrict__ state,       // ws: h[40] then c[40]
    const float* __restrict__ W_lin, // [40]
    const float* __restrict__ b_lin, // [1]
    float* __restrict__ out,         // [1]
    int chunkT, int first, int last)
{
    __shared__ __align__(16) float h_s[LSTM_H];
    __shared__ __align__(16) float c_s[LSTM_H];
    __shared__ __align__(16) float gate_s[LSTM_G];

    const int j = threadIdx.x;

    // Pin W_hh row j in VGPRs.
    float w[LSTM_H];
#pragma unroll
    for (int k = 0; k < LSTM_H; ++k) w[k] = W_hh[(size_t)j * LSTM_H + k];

    if (j < LSTM_H) {
        h_s[j] = first ? 0.0f : state[j];
        c_s[j] = first ? 0.0f : state[LSTM_H + j];
    }
    __syncthreads();

    const bool is_g = (j >= 2 * LSTM_H) && (j < 3 * LSTM_H);
    const float4* h4 = (const float4*)h_s;

    float xv = xp[j];  // software-pipelined xp load
    for (int t = 0; t < chunkT; ++t) {
        // ---- recurrent matvec: 10 x ds_load_b128 + 40 FMA, 4 accumulators
        float acc0 = xv, acc1 = 0.0f, acc2 = 0.0f, acc3 = 0.0f;
#pragma unroll
        for (int k4 = 0; k4 < LSTM_H / 4; ++k4) {
            float4 hv = h4[k4];
            acc0 += w[4 * k4 + 0] * hv.x;
            acc1 += w[4 * k4 + 1] * hv.y;
            acc2 += w[4 * k4 + 2] * hv.z;
            acc3 += w[4 * k4 + 3] * hv.w;
        }
        // Issue next timestep's xp load now (branch-free, clamped index);
        // it is consumed only on the next iteration -> full step of slack.
        const int tn = (t + 1 < chunkT) ? (t + 1) : t;
        float xnext = xp[(size_t)tn * LSTM_G + j];

        float z = (acc0 + acc1) + (acc2 + acc3);
        gate_s[j] = is_g ? fast_tanh(z) : fast_sigmoid(z);
        __syncthreads();

        if (j < LSTM_H) {
            float gi = gate_s[j];
            float gf = gate_s[LSTM_H + j];
            float gg = gate_s[2 * LSTM_H + j];
            float go = gate_s[3 * LSTM_H + j];
            float c = gf * c_s[j] + gi * gg;
            c_s[j] = c;
            h_s[j] = go * fast_tanh(c);
        }
        __syncthreads();
        xv = xnext;
    }

    if (j < LSTM_H) {
        state[j] = h_s[j];
        state[LSTM_H + j] = c_s[j];
    }
    if (last && j == 0) {
        float p = b_lin[0];
#pragma unroll
        for (int m = 0; m < LSTM_H; ++m) p += h_s[m] * W_lin[m];
        out[0] = p;
    }
}

// ---------------------------------------------------------------------------
// Fallback: fully fused scan (projection computed inline, no workspace).
// Used only if d_ws is too small for even a 16-step x_proj chunk.
// ---------------------------------------------------------------------------
__global__ __launch_bounds__(LSTM_G) void lstm_fused_kernel(
    const float* __restrict__ x,     // [T, 6]
    const float* __restrict__ W_ih,  // [160, 6]
    const float* __restrict__ W_hh,  // [160, 40]
    const float* __restrict__ b_ih,
    const float* __restrict__ b_hh,
    const float* __restrict__ W_lin,
    const float* __restrict__ b_lin,
    float* __restrict__ out, int T)
{
    __shared__ __align__(16) float h_s[LSTM_H];
    __shared__ __align__(16) float c_s[LSTM_H];
    __shared__ __align__(16) float gate_s[LSTM_G];

    const int j = threadIdx.x;
    float wi[LSTM_IN];
#pragma unroll
    for (int k = 0; k < LSTM_IN; ++k) wi[k] = W_ih[(size_t)j * LSTM_IN + k];
    const float bias = b_ih[j] + b_hh[j];
    float w[LSTM_H];
#pragma unroll
    for (int k = 0; k < LSTM_H; ++k) w[k] = W_hh[(size_t)j * LSTM_H + k];

    if (j < LSTM_H) { h_s[j] = 0.0f; c_s[j] = 0.0f; }
    __syncthreads();

    const bool is_g = (j >= 2 * LSTM_H) && (j < 3 * LSTM_H);
    const float4* h4 = (const float4*)h_s;

    for (int t = 0; t < T; ++t) {
        const float* xr = x + (size_t)t * LSTM_IN;
        float acc0 = bias, acc1 = 0.0f, acc2 = 0.0f, acc3 = 0.0f;
#pragma unroll
        for (int k = 0; k < LSTM_IN; ++k) acc1 += wi[k] * xr[k];
#pragma unroll
        for (int k4 = 0; k4 < LSTM_H / 4; ++k4) {
            float4 hv = h4[k4];
            acc0 += w[4 * k4 + 0] * hv.x;
            acc1 += w[4 * k4 + 1] * hv.y;
            acc2 += w[4 * k4 + 2] * hv.z;
            acc3 += w[4 * k4 + 3] * hv.w;
        }
        float z = (acc0 + acc1) + (acc2 + acc3);
        gate_s[j] = is_g ? fast_tanh(z) : fast_sigmoid(z);
        __syncthreads();
        if (j < LSTM_H) {
            float gi = gate_s[j];
            float gf = gate_s[LSTM_H + j];
            float gg = gate_s[2 * LSTM_H + j];
            float go = gate_s[3 * LSTM_H + j];
            float c = gf * c_s[j] + gi * gg;
            c_s[j] = c;
            h_s[j] = go * fast_tanh(c);
        }
        __syncthreads();
    }
    if (j == 0) {
        float p = b_lin[0];
#pragma unroll
        for (int m = 0; m < LSTM_H; ++m) p += h_s[m] * W_lin[m];
        out[0] = p;
    }
}

// ---------------------------------------------------------------------------
extern "C" void kernel_launch(void* const* d_in, const int* in_sizes, int n_in,
                              void* d_out, int out_size, void* d_ws, size_t ws_size,
                              hipStream_t stream) {
    const float* x     = (const float*)d_in[0];  // [T, 6]
    const float* W_ih  = (const float*)d_in[1];  // [160, 6]
    const float* W_hh  = (const float*)d_in[2];  // [160, 40]
    const float* b_ih  = (const float*)d_in[3];  // [160]
    const float* b_hh  = (const float*)d_in[4];  // [160]
    const float* W_lin = (const float*)d_in[5];  // [1, 40]
    const float* b_lin = (const float*)d_in[6];  // [1]
    float* out = (float*)d_out;

    const int T = in_sizes[0] / LSTM_IN;         // 262144

    // Workspace: 256B state (h,c) + x_proj chunk. Pick the largest power-of-2
    // chunk (<=8192, >=16, dividing T) whose x_proj tile fits in d_ws; an
    // 8192-step tile is 5.2MB -> L2-resident between producer and consumer.
    const size_t stateBytes = 256;
    int chunk = 8192;
    while (chunk > 16 &&
           (stateBytes + (size_t)chunk * LSTM_G * sizeof(float) > ws_size ||
            (T % chunk) != 0)) {
        chunk >>= 1;
    }
    const bool useWmma =
        d_ws != nullptr &&
        (stateBytes + (size_t)chunk * LSTM_G * sizeof(float) <= ws_size) &&
        (T % chunk) == 0;

    if (useWmma) {
        float* state = (float*)d_ws;
        float* xp = (float*)((char*)d_ws + stateBytes);
        const int nChunks = T / chunk;
        for (int ci = 0; ci < nChunks; ++ci) {
            const int t0 = ci * chunk;
            xproj_wmma_kernel<<<(chunk / 16) * 10, 32, 0, stream>>>(
                x + (size_t)t0 * LSTM_IN, W_ih, b_ih, b_hh, xp, chunk);
            lstm_scan_kernel<<<1, LSTM_G, 0, stream>>>(
                xp, W_hh, state, W_lin, b_lin, out, chunk,
                ci == 0 ? 1 : 0, ci == nChunks - 1 ? 1 : 0);
        }
    } else {
        lstm_fused_kernel<<<1, LSTM_G, 0, stream>>>(
            x, W_ih, W_hh, b_ih, b_hh, W_lin, b_lin, out, T);
    }
}